// HybridQuantumNAT_65481071400264
// MI455X (gfx1250) — compile-verified
//
#include <hip/hip_runtime.h>
#include <hip/hip_bf16.h>
#include <math.h>

typedef __attribute__((ext_vector_type(2))) float v2f;
typedef __attribute__((ext_vector_type(8))) float v8f;
typedef int v4i __attribute__((vector_size(16)));   // matches builtin param type

#define BATCH 8192
#define SPB 16            // samples per block in fused CNN kernel
#define BN_EPS 1e-5f
#define THRESHOLD 0.9f
#define JCHUNKS 8         // j-loop split factor in k_adj (occupancy)

// ---------------------------------------------------------------------------
// K0: zero the small accumulator header in workspace
// ---------------------------------------------------------------------------
__global__ void k_zero(float* __restrict__ acc) {
    int t = threadIdx.x;
    if (t < 16) acc[t] = 0.0f;
}

// ---------------------------------------------------------------------------
// K1: fused CNN: conv1+relu+pool -> conv2+relu+pool -> fc1(WMMA) -> fc2
// Block = 128 threads (4 waves), owns 16 samples.  ~157KB LDS -> 2 blocks/WGP.
// x is staged into LDS with GLOBAL_LOAD_ASYNC_TO_LDS_B128 when available.
// fc1 uses V_WMMA_F32_16X16X4_F32: A = flat[16 x 784] tile, B = fc1_w^T tile.
// ---------------------------------------------------------------------------
__global__ __launch_bounds__(128)
void k_cnn(const float* __restrict__ x,
           const float* __restrict__ w1, const float* __restrict__ b1,
           const float* __restrict__ w2, const float* __restrict__ b2,
           const float* __restrict__ fw1, const float* __restrict__ fb1,
           const float* __restrict__ fw2, const float* __restrict__ fb2,
           float* __restrict__ proj_raw,
           float* __restrict__ sums, float* __restrict__ sumsq)
{
    __shared__ float sA [SPB * 784];        // x image, later the flattened conv2 output
    __shared__ float sH1[SPB * 8 * 196];    // pooled conv1 output [s][c][14*14]
    __shared__ float sFC[SPB * 64];         // fc1 output
    __shared__ float sW1[72],   sB1[8];
    __shared__ float sW2[1152], sB2[16];
    __shared__ float sW3[256],  sB3[4];

    const int tid   = threadIdx.x;
    const int sbase = blockIdx.x * SPB;

    for (int i = tid; i < 72;   i += 128) sW1[i] = w1[i];
    for (int i = tid; i < 8;    i += 128) sB1[i] = b1[i];
    for (int i = tid; i < 1152; i += 128) sW2[i] = w2[i];
    for (int i = tid; i < 16;   i += 128) sB2[i] = b2[i];
    for (int i = tid; i < 256;  i += 128) sW3[i] = fw2[i];
    for (int i = tid; i < 4;    i += 128) sB3[i] = fb2[i];

    // --- stage x[sbase .. sbase+15] into LDS ---------------------------------
#if __has_builtin(__builtin_amdgcn_global_load_async_to_lds_b128)
    {
        v4i* gx = (v4i*)(x + sbase * 784);   // global source, 16B granules
        v4i* ls = (v4i*)sA;                  // LDS destination
        // 16 * 784 * 4B = 50176B = 3136 x b128 async DMA transfers into LDS
        for (int i = tid; i < SPB * 784 / 4; i += 128)
            __builtin_amdgcn_global_load_async_to_lds_b128(gx + i, ls + i, 0, 0);
    }
#if __has_builtin(__builtin_amdgcn_s_wait_asynccnt)
    __builtin_amdgcn_s_wait_asynccnt(0);
#else
    asm volatile("s_wait_asynccnt 0" ::: "memory");
#endif
#else
    for (int i = tid; i < SPB * 784; i += 128) sA[i] = x[sbase * 784 + i];
#endif
    __syncthreads();

    // conv1 (1->8, 3x3 SAME) + maxpool2 (max of conv then relu == relu then pool)
    for (int item = tid; item < SPB * 8 * 196; item += 128) {
        int s   = item / (8 * 196);
        int rem = item % (8 * 196);
        int c   = rem / 196;
        int p   = rem % 196;
        int pi  = p / 14, pj = p % 14;
        const float* xi = &sA[s * 784];
        const float* wc = &sW1[c * 9];
        float m = -1e30f;
        for (int di = 0; di < 2; di++)
        for (int dj = 0; dj < 2; dj++) {
            int ri = 2 * pi + di, rj = 2 * pj + dj;
            float acc = sB1[c];
            for (int u = 0; u < 3; u++) {
                int yy = ri + u - 1;
                if (yy < 0 || yy > 27) continue;
                for (int v = 0; v < 3; v++) {
                    int xx = rj + v - 1;
                    if (xx < 0 || xx > 27) continue;
                    acc = fmaf(xi[yy * 28 + xx], wc[u * 3 + v], acc);
                }
            }
            m = fmaxf(m, acc);
        }
        sH1[(s * 8 + c) * 196 + p] = fmaxf(m, 0.0f);
    }
    __syncthreads();

    // conv2 (8->16, 3x3 SAME) + maxpool2, write flattened result over sA
    for (int item = tid; item < SPB * 16 * 49; item += 128) {
        int s   = item / (16 * 49);
        int rem = item % (16 * 49);
        int c   = rem / 49;
        int p   = rem % 49;
        int pi  = p / 7, pj = p % 7;
        float m = -1e30f;
        for (int di = 0; di < 2; di++)
        for (int dj = 0; dj < 2; dj++) {
            int ri = 2 * pi + di, rj = 2 * pj + dj;
            float acc = sB2[c];
            for (int ic = 0; ic < 8; ic++) {
                const float* hp = &sH1[(s * 8 + ic) * 196];
                const float* wc = &sW2[(c * 8 + ic) * 9];
                for (int u = 0; u < 3; u++) {
                    int yy = ri + u - 1;
                    if (yy < 0 || yy > 13) continue;
                    for (int v = 0; v < 3; v++) {
                        int xx = rj + v - 1;
                        if (xx < 0 || xx > 13) continue;
                        acc = fmaf(hp[yy * 14 + xx], wc[u * 3 + v], acc);
                    }
                }
            }
            m = fmaxf(m, acc);
        }
        sA[s * 784 + c * 49 + p] = fmaxf(m, 0.0f);   // flat layout [s][c*49+p]
    }
    __syncthreads();

    // fc1 via WMMA f32 16x16x4.  M = 16 samples, N = 64 (4 waves x 16), K = 784.
    {
        const int lane  = tid & 31;
        const int wv    = tid >> 5;          // 0..3 -> N tile
        const int lm    = lane & 15;         // M row (A) / N col (B)
        const int khalf = lane >> 4;         // K sub-pair select
        const int n0    = wv * 16;
        const v2f* fw1v = (const v2f*)fw1;   // fc1_w rows are 784 floats (v2f-aligned)
        v8f cacc = {};
        for (int kk = 0; kk < 784; kk += 4) {
            v2f a, b;
            a[0] = sA[lm * 784 + kk + 2 * khalf];
            a[1] = sA[lm * 784 + kk + 2 * khalf + 1];
            b    = fw1v[(n0 + lm) * 392 + kk / 2 + khalf];   // B[k][n] = fc1_w[n][k]
            cacc = __builtin_amdgcn_wmma_f32_16x16x4_f32(
                false, a, false, b, (short)0, cacc, false, false);
        }
        // C layout: VGPR r, lane L -> (M = r + 8*(L/16), N = n0 + L%16)
        const int   n    = n0 + lm;
        const float bias = fb1[n];
        for (int r = 0; r < 8; r++) {
            int m = r + 8 * khalf;
            sFC[m * 64 + n] = fmaxf(cacc[r] + bias, 0.0f);
        }
    }
    __syncthreads();

    // fc2: 16x4 output, 64-wide dot per element; also BN partial sums
    if (tid < 64) {
        int m = tid >> 2, cc = tid & 3;
        float acc = sB3[cc];
        for (int k = 0; k < 64; k++)
            acc = fmaf(sFC[m * 64 + k], sW3[cc * 64 + k], acc);
        proj_raw[(sbase + m) * 4 + cc] = acc;
        atomicAdd(&sums[cc],  acc);
        atomicAdd(&sumsq[cc], acc * acc);
    }
}

// ---------------------------------------------------------------------------
// K2: batch-norm (training-mode batch stats, biased var) + row normalize.
// Also zeroes wp for K3's atomic accumulation.
// ---------------------------------------------------------------------------
__global__ void k_bn(const float* __restrict__ proj_raw,
                     const float* __restrict__ sums, const float* __restrict__ sumsq,
                     const float* __restrict__ gamma, const float* __restrict__ beta,
                     float* __restrict__ proj, float* __restrict__ pn,
                     float* __restrict__ wp)
{
    int b = blockIdx.x * blockDim.x + threadIdx.x;
    if (b >= BATCH) return;
    float p[4];
    float ss = 0.0f;
    for (int c = 0; c < 4; c++) {
        float mean = sums[c]  * (1.0f / BATCH);
        float var  = sumsq[c] * (1.0f / BATCH) - mean * mean;
        float v = (proj_raw[b * 4 + c] - mean) * rsqrtf(var + BN_EPS) * gamma[c] + beta[c];
        p[c] = v;
        ss  += v * v;
        proj[b * 4 + c] = v;
        wp[b * 4 + c]   = 0.0f;
    }
    float inv = 1.0f / (sqrtf(ss) + 1e-12f);
    for (int c = 0; c < 4; c++) pn[b * 4 + c] = p[c] * inv;
}

// ---------------------------------------------------------------------------
// K3: O(B^2) fidelity adjacency + weighted_proj = adj @ proj.
// One wave per (16-row i-tile, j-chunk): 512 tiles x 8 chunks = 4096 waves.
// Per j-tile: one V_WMMA_F32_16X16X4_F32 gives the 16x16 dot tile; branchless
// threshold; per-lane accumulate adj@proj; shuffle-reduce; atomicAdd into wp.
// ---------------------------------------------------------------------------
__global__ __launch_bounds__(128)
void k_adj(const float* __restrict__ pn, const float* __restrict__ proj,
           float* __restrict__ wp)
{
    const int lane  = threadIdx.x & 31;
    const int w     = blockIdx.x * 4 + (threadIdx.x >> 5);
    const int itile = w >> 3;           // 0..511
    const int chunk = w & (JCHUNKS - 1);
    const int ibase = itile * 16;
    const int lm    = lane & 15;
    const int khalf = lane >> 4;

    const v2f*    pn2   = (const v2f*)pn;
    const float4* proj4 = (const float4*)proj;

    // A fragment: Pn_i, 16x4, loaded once (pn is L2-resident, 128 KB)
    v2f a = pn2[(ibase + lm) * 2 + khalf];

    float wpp[8][4];
    for (int r = 0; r < 8; r++)
        for (int c = 0; c < 4; c++) wpp[r][c] = 0.0f;

    const int jtiles = (BATCH / 16) / JCHUNKS;   // 64
    const int jt0    = chunk * jtiles;
    for (int jt = jt0; jt < jt0 + jtiles; jt++) {
        const int jbase = jt * 16;
        v2f    b  = pn2[(jbase + lm) * 2 + khalf];   // B: Pn_j^T (same pattern as A)
        float4 pj = proj4[jbase + lm];               // proj row for this lane's N col

        v8f z = {};
        v8f d = __builtin_amdgcn_wmma_f32_16x16x4_f32(
            false, a, false, b, (short)0, z, false, false);

        const int gj = jbase + lm;
        for (int r = 0; r < 8; r++) {
            float dot = d[r];
            float fid = dot * dot;
            int   gi  = ibase + r + 8 * khalf;
            float msk = (fid >= THRESHOLD && gi != gj) ? 1.0f : 0.0f;  // branchless
            wpp[r][0] = fmaf(msk, pj.x, wpp[r][0]);
            wpp[r][1] = fmaf(msk, pj.y, wpp[r][1]);
            wpp[r][2] = fmaf(msk, pj.z, wpp[r][2]);
            wpp[r][3] = fmaf(msk, pj.w, wpp[r][3]);
        }
    }

    // reduce across the 16 lanes of each half (xor masks < 16 keep the half)
    for (int r = 0; r < 8; r++)
        for (int c = 0; c < 4; c++) {
            float v = wpp[r][c];
            v += __shfl_xor(v, 8, 32);
            v += __shfl_xor(v, 4, 32);
            v += __shfl_xor(v, 2, 32);
            v += __shfl_xor(v, 1, 32);
            wpp[r][c] = v;
        }
    if (lm == 0) {
        int mrow = ibase + 8 * khalf;   // lane 0 -> rows +0..7, lane 16 -> +8..15
        for (int r = 0; r < 8; r++)
            for (int c = 0; c < 4; c++)
                atomicAdd(&wp[(mrow + r) * 4 + c], wpp[r][c]);
    }
}

// ---------------------------------------------------------------------------
// K4: ksum = sum((proj - wp)^2) via wave reduce + atomics
// ---------------------------------------------------------------------------
__global__ void k_kval(const float* __restrict__ proj, const float* __restrict__ wp,
                       float* __restrict__ ksum)
{
    int i = blockIdx.x * blockDim.x + threadIdx.x;
    float d = 0.0f;
    if (i < BATCH * 4) {
        float t = proj[i] - wp[i];
        d = t * t;
    }
    d += __shfl_xor(d, 16, 32);
    d += __shfl_xor(d, 8, 32);
    d += __shfl_xor(d, 4, 32);
    d += __shfl_xor(d, 2, 32);
    d += __shfl_xor(d, 1, 32);
    if ((threadIdx.x & 31) == 0) atomicAdd(ksum, d);
}

// ---------------------------------------------------------------------------
// K5: logits -> sigmoid -> [B,2] output (probs, 1-probs)
// ---------------------------------------------------------------------------
__global__ void k_out(const float* __restrict__ proj, const float* __restrict__ wp,
                      const float* __restrict__ ksum, float* __restrict__ out)
{
    int b = blockIdx.x * blockDim.x + threadIdx.x;
    if (b >= BATCH) return;
    float k = expf(-(*ksum));   // GAMMA = 1
    float s = 0.0f;
    for (int c = 0; c < 4; c++) s += proj[b * 4 + c] + wp[b * 4 + c];
    float logit = s + k;        // SHIFT = 0
    float p = 1.0f / (1.0f + expf(-logit));
    out[2 * b]     = p;
    out[2 * b + 1] = 1.0f - p;
}

// ---------------------------------------------------------------------------
extern "C" void kernel_launch(void* const* d_in, const int* in_sizes, int n_in,
                              void* d_out, int out_size, void* d_ws, size_t ws_size,
                              hipStream_t stream)
{
    const float* x       = (const float*)d_in[0];
    const float* conv1_w = (const float*)d_in[1];
    const float* conv1_b = (const float*)d_in[2];
    const float* conv2_w = (const float*)d_in[3];
    const float* conv2_b = (const float*)d_in[4];
    const float* fc1_w   = (const float*)d_in[5];
    const float* fc1_b   = (const float*)d_in[6];
    const float* fc2_w   = (const float*)d_in[7];
    const float* fc2_b   = (const float*)d_in[8];
    const float* bn_g    = (const float*)d_in[9];
    const float* bn_b    = (const float*)d_in[10];
    float* out = (float*)d_out;

    // workspace layout (floats): [0..3] sums, [4..7] sumsq, [8] ksum, pad to 16,
    // then proj_raw / proj / pn / wp, each BATCH*4.  Total ~512 KB.
    float* W        = (float*)d_ws;
    float* sums     = W;
    float* sumsq    = W + 4;
    float* ksum     = W + 8;
    float* proj_raw = W + 16;
    float* proj     = proj_raw + BATCH * 4;
    float* pn       = proj     + BATCH * 4;
    float* wp       = pn       + BATCH * 4;

    k_zero<<<1, 32, 0, stream>>>(W);
    k_cnn <<<BATCH / SPB, 128, 0, stream>>>(x, conv1_w, conv1_b, conv2_w, conv2_b,
                                            fc1_w, fc1_b, fc2_w, fc2_b,
                                            proj_raw, sums, sumsq);
    k_bn  <<<BATCH / 256, 256, 0, stream>>>(proj_raw, sums, sumsq, bn_g, bn_b,
                                            proj, pn, wp);
    k_adj <<<(BATCH / 16) * JCHUNKS / 4, 128, 0, stream>>>(pn, proj, wp);
    k_kval<<<(BATCH * 4 + 255) / 256, 256, 0, stream>>>(proj, wp, ksum);
    k_out <<<BATCH / 256, 256, 0, stream>>>(proj, wp, ksum, out);
}